// DINA_25503515804209
// MI455X (gfx1250) — compile-verified
//
#include <hip/hip_runtime.h>
#include <hip/hip_bf16.h>

typedef __attribute__((ext_vector_type(2))) float v2f;
typedef __attribute__((ext_vector_type(8))) float v8f;
typedef __attribute__((ext_vector_type(4))) int   v4i;

#define B_   8
#define L1_  1200
#define D1_  64
#define L2_  848
#define D2_  48
#define H_   2
#define C_   64
#define Q_   2048            // L1_ + L2_ == 2048 exactly
#define SCALE_ 0.125f        // 64^-0.5
#define BLKT 53              // 848/16 masked 16x16 tiles

// workspace layout (float offsets)
#define OFF_MPAD  0ull
#define OFF_Y     (OFF_MPAD + (size_t)B_*Q_*C_)
#define OFF_ROWMX (OFF_Y + (size_t)B_*H_*Q_*C_)
#define OFF_COLMX (OFF_ROWMX + (size_t)B_*H_*Q_)
#define OFF_ALPHA (OFF_COLMX + (size_t)B_*H_*Q_)
#define OFF_KMIN  (OFF_ALPHA + (size_t)B_*H_*Q_)

// ---------------- init reduction buffers ----------------
__global__ void k0_init(unsigned* rowmx, unsigned* colmx, unsigned* kmin) {
  int i = blockIdx.x * blockDim.x + threadIdx.x;
  if (i < B_*H_*Q_) { rowmx[i] = 0u; colmx[i] = 0u; }   // 0.0f bits; all candidates >= 0
  if (i == 0) kmin[0] = 0x7F7FFFFFu;                    // FLT_MAX bits
}

// ---------------- pad + concat into Mpad (B,2048,64) ----------------
__global__ void k1_pad(const float* __restrict__ x1, const float* __restrict__ x2,
                       float* __restrict__ Mpad) {
  int i = blockIdx.x * blockDim.x + threadIdx.x;   // < B*Q*C = 2^20
  int ci = i & 63;
  int qi = (i >> 6) & (Q_ - 1);
  int b  = i >> 17;
  float v;
  if (qi < L1_) {
    v = x1[((size_t)b * L1_ + qi) * D1_ + ci];
  } else {
    int q2 = qi - L1_;
    v = (ci < D2_) ? x2[((size_t)b * L2_ + q2) * D2_ + ci] : 0.0f;
  }
  Mpad[i] = v;
}

// ---------------- Y[bh] = Mpad[b] @ U[h]  (2048x64 @ 64x64) ----------------
__global__ void k2_mu(const float* __restrict__ Mpad, const float* __restrict__ U,
                      float* __restrict__ Y) {
  int wave = threadIdx.x >> 5;
  int lane = threadIdx.x & 31;
  int t  = blockIdx.x * 8 + wave;      // < 16*128*4 = 8192
  int bh = t >> 9;
  int rem = t & 511;
  int tm = rem >> 2;
  int tn = rem & 3;
  int b = bh >> 1, h = bh & 1;
  int half = lane >> 4, mr = lane & 15;

  const float* A  = Mpad + ((size_t)b * Q_ + tm * 16 + mr) * C_;
  const float* Ub = U + (size_t)h * C_ * C_;

  v8f acc = {};
#pragma unroll
  for (int k0 = 0; k0 < 64; k0 += 4) {
    int ka = k0 + 2 * half;
    v2f a = *(const v2f*)(A + ka);
    v2f bb;
    bb.x = Ub[(size_t)ka * C_ + tn * 16 + mr];
    bb.y = Ub[(size_t)(ka + 1) * C_ + tn * 16 + mr];
    acc = __builtin_amdgcn_wmma_f32_16x16x4_f32(false, a, false, bb,
                                                (short)0, acc, false, false);
  }
  float* Yo = Y + ((size_t)bh * Q_ + tm * 16) * C_ + tn * 16 + mr;
#pragma unroll
  for (int r = 0; r < 8; ++r)
    Yo[(size_t)(r + 8 * half) * C_] = acc[r];
}

// ---------------- k3: per block, 128x128 super-tile of S = relu(scale*Y@Mpad^T) ----------------
// LDS staging with XOR-quad swizzle; fused row/col max + head-0 global min reductions.
#define XR(x) { x = fmaxf(x, __shfl_xor(x, 1, 32)); x = fmaxf(x, __shfl_xor(x, 2, 32)); \
                x = fmaxf(x, __shfl_xor(x, 4, 32)); x = fmaxf(x, __shfl_xor(x, 8, 32)); }

__device__ __forceinline__ void stage_b128(const float* __restrict__ g, float* l) {
#if defined(__has_builtin)
#if __has_builtin(__builtin_amdgcn_global_load_async_to_lds_b128)
  typedef __attribute__((address_space(1))) v4i* gp_t;   // global v4i*
  typedef __attribute__((address_space(3))) v4i* lp_t;   // LDS v4i*
  __builtin_amdgcn_global_load_async_to_lds_b128((gp_t)(v4i*)g, (lp_t)(v4i*)l, 0, 0);
  return;
#endif
#endif
  *(float4*)l = *(const float4*)g;
}

__device__ __forceinline__ void wait_async_then_barrier() {
#if defined(__has_builtin)
#if __has_builtin(__builtin_amdgcn_global_load_async_to_lds_b128)
#if __has_builtin(__builtin_amdgcn_s_wait_asynccnt)
  __builtin_amdgcn_s_wait_asynccnt(0);
#else
  asm volatile("s_wait_asynccnt 0x0" ::: "memory");
#endif
#endif
#endif
  __syncthreads();
}

__global__ __launch_bounds__(256) void k3_gemm(
    const float* __restrict__ Mpad, const float* __restrict__ Y,
    unsigned* __restrict__ rowmx, unsigned* __restrict__ colmx,
    unsigned* __restrict__ kmin) {
  __shared__ __align__(16) float As[128 * 64];
  __shared__ __align__(16) float Bs[128 * 64];

  int tid  = threadIdx.x;
  int wave = tid >> 5, lane = tid & 31;
  int half = lane >> 4, mr = lane & 15;

  int blk = blockIdx.x;            // bh*256 + tmS*16 + tnS  (16*16*16 = 4096 blocks)
  int bh  = blk >> 8;
  int tmS = (blk >> 4) & 15;
  int tnS = blk & 15;
  int b = bh >> 1, h = bh & 1;

  const float* Ag = Y    + ((size_t)bh * Q_ + tmS * 128) * C_;  // 128x64, contiguous 32KB
  const float* Bg = Mpad + ((size_t)b  * Q_ + tnS * 128) * C_;  // 128x64, contiguous 32KB

  // stage both strips; float4 (row,q) -> LDS float idx row*64 + ((q^(row&15))<<2)
#pragma unroll
  for (int i = 0; i < 8; ++i) {
    int f = tid + i * 256;                       // float4 index in 128x16
    int row = f >> 4, q = f & 15;
    int dst = row * 64 + ((q ^ (row & 15)) << 2);
    stage_b128(Ag + (size_t)f * 4, As + dst);
    stage_b128(Bg + (size_t)f * 4, Bs + dst);
  }
  wait_async_then_barrier();

  // wave computes a 16x128 strip: 8 N-tiles, A fragment reused 8x per k-step
  v8f acc[8];
  v8f zero = {};
#pragma unroll
  for (int nt = 0; nt < 8; ++nt) acc[nt] = zero;

  int arow = wave * 16 + mr;
#pragma unroll
  for (int k0 = 0; k0 < 64; k0 += 4) {
    int sw = (((k0 >> 2) ^ mr) << 2) + 2 * half;   // swizzled k-offset within a row
    v2f a = *(const v2f*)(As + arow * 64 + sw);
#pragma unroll
    for (int nt = 0; nt < 8; ++nt) {
      v2f bb = *(const v2f*)(Bs + (nt * 16 + mr) * 64 + sw);
      acc[nt] = __builtin_amdgcn_wmma_f32_16x16x4_f32(false, a, false, bb,
                                                      (short)0, acc[nt], false, false);
    }
  }

  // fused reductions: K = relu(scale*S); masked 16x16 tiles excluded from max
  int rowTile = tmS * 8 + wave;
  bool rowInBlk = rowTile < BLKT;
  float rm[8];
#pragma unroll
  for (int j = 0; j < 8; ++j) rm[j] = 0.f;       // safe: true maxes are >= 0 (ReLU)
  float gmin = 3.4e38f;

#pragma unroll
  for (int nt = 0; nt < 8; ++nt) {
    float t0 = fmaxf(acc[nt][0] * SCALE_, 0.f), t1 = fmaxf(acc[nt][1] * SCALE_, 0.f);
    float t2 = fmaxf(acc[nt][2] * SCALE_, 0.f), t3 = fmaxf(acc[nt][3] * SCALE_, 0.f);
    float t4 = fmaxf(acc[nt][4] * SCALE_, 0.f), t5 = fmaxf(acc[nt][5] * SCALE_, 0.f);
    float t6 = fmaxf(acc[nt][6] * SCALE_, 0.f), t7 = fmaxf(acc[nt][7] * SCALE_, 0.f);

    if (h == 0) {                                // pre-mask global min over head 0
      float mn = fminf(fminf(fminf(t0, t1), fminf(t2, t3)),
                       fminf(fminf(t4, t5), fminf(t6, t7)));
      gmin = fminf(gmin, mn);
    }

    int colTile = tnS * 8 + nt;
    bool msk = rowInBlk && (colTile < BLKT);
    if (!msk) {
      float cm = fmaxf(fmaxf(fmaxf(t0, t1), fmaxf(t2, t3)),
                       fmaxf(fmaxf(t4, t5), fmaxf(t6, t7)));
      cm = fmaxf(cm, __shfl_xor(cm, 16, 32));
      if (half == 0)
        atomicMax(&colmx[(size_t)bh * Q_ + colTile * 16 + mr], __float_as_uint(cm));
      rm[0] = fmaxf(rm[0], t0); rm[1] = fmaxf(rm[1], t1);
      rm[2] = fmaxf(rm[2], t2); rm[3] = fmaxf(rm[3], t3);
      rm[4] = fmaxf(rm[4], t4); rm[5] = fmaxf(rm[5], t5);
      rm[6] = fmaxf(rm[6], t6); rm[7] = fmaxf(rm[7], t7);
    }
  }

  if (h == 0) {
    gmin = fminf(gmin, __shfl_xor(gmin, 16, 32));
    gmin = fminf(gmin, __shfl_xor(gmin,  8, 32));
    gmin = fminf(gmin, __shfl_xor(gmin,  4, 32));
    gmin = fminf(gmin, __shfl_xor(gmin,  2, 32));
    gmin = fminf(gmin, __shfl_xor(gmin,  1, 32));
    if (lane == 0) atomicMin(kmin, __float_as_uint(gmin));
  }

  XR(rm[0]) XR(rm[1]) XR(rm[2]) XR(rm[3]) XR(rm[4]) XR(rm[5]) XR(rm[6]) XR(rm[7])
  int rr = mr;
  if (rr < 8) {
    float val = rm[0];
    val = (rr == 1) ? rm[1] : val; val = (rr == 2) ? rm[2] : val;
    val = (rr == 3) ? rm[3] : val; val = (rr == 4) ? rm[4] : val;
    val = (rr == 5) ? rm[5] : val; val = (rr == 6) ? rm[6] : val;
    val = (rr == 7) ? rm[7] : val;
    atomicMax(&rowmx[(size_t)bh * Q_ + rowTile * 16 + rr + 8 * half],
              __float_as_uint(val));
  }
}

// ---------------- alpha = rows + cols with min_val fill on masked block ----------------
__global__ void k4_alpha(const unsigned* __restrict__ rowmx, const unsigned* __restrict__ colmx,
                         const unsigned* __restrict__ kmin, float* __restrict__ alpha) {
  int i = blockIdx.x * blockDim.x + threadIdx.x;   // < B*H*Q
  int qi = i & (Q_ - 1);
  float mv = __uint_as_float(kmin[0]);
  float rw = __uint_as_float(rowmx[i]);
  float cl = __uint_as_float(colmx[i]);
  if (qi < L2_) { rw = fmaxf(rw, mv); cl = fmaxf(cl, mv); }
  alpha[i] = rw + cl;
}

// ---------------- segment softmax + weighted aggregation ----------------
__global__ void k5_agg(const float* __restrict__ Mpad, const float* __restrict__ alpha,
                       float* __restrict__ out) {
  __shared__ float red[256];
  int tid = threadIdx.x;
  int idx = blockIdx.x;                 // 32 blocks
  int b = idx >> 2, h = (idx >> 1) & 1, seg = idx & 1;
  int base = seg ? L1_ : 0;
  int len  = seg ? L2_ : L1_;
  const float* al = alpha + ((size_t)(b * H_ + h)) * Q_ + base;
  const float* Mr = Mpad + ((size_t)b * Q_ + base) * C_;

  float am = -3.4e38f;
  for (int l = tid; l < len; l += 256) am = fmaxf(am, al[l]);
  red[tid] = am; __syncthreads();
  for (int s = 128; s > 0; s >>= 1) {
    if (tid < s) red[tid] = fmaxf(red[tid], red[tid + s]);
    __syncthreads();
  }
  am = red[0]; __syncthreads();

  float zs = 0.f;
  for (int l = tid; l < len; l += 256) zs += expf(al[l] - am);
  red[tid] = zs; __syncthreads();
  for (int s = 128; s > 0; s >>= 1) {
    if (tid < s) red[tid] += red[tid + s];
    __syncthreads();
  }
  float Z = red[0]; __syncthreads();

  int c = tid & 63;
  float acc = 0.f;
  for (int l = tid >> 6; l < len; l += 4)
    acc += expf(al[l] - am) * Mr[(size_t)l * C_ + c];
  red[tid] = acc; __syncthreads();
  if (tid < 64) {
    float s = red[tid] + red[tid + 64] + red[tid + 128] + red[tid + 192];
    out[(size_t)seg * (B_ * H_ * C_) + (b * H_ + h) * C_ + c] = s / Z;
  }
}

extern "C" void kernel_launch(void* const* d_in, const int* in_sizes, int n_in,
                              void* d_out, int out_size, void* d_ws, size_t ws_size,
                              hipStream_t stream) {
  const float* x1 = (const float*)d_in[0];
  const float* x2 = (const float*)d_in[1];
  const float* U  = (const float*)d_in[2];
  float* ws = (float*)d_ws;

  float*    Mpad  = ws + OFF_MPAD;
  float*    Y     = ws + OFF_Y;
  unsigned* rowmx = (unsigned*)(ws + OFF_ROWMX);
  unsigned* colmx = (unsigned*)(ws + OFF_COLMX);
  float*    alpha = ws + OFF_ALPHA;
  unsigned* kmin  = (unsigned*)(ws + OFF_KMIN);

  k0_init <<<(B_*H_*Q_ + 255) / 256, 256, 0, stream>>>(rowmx, colmx, kmin);
  k1_pad  <<<(B_*Q_*C_) / 256, 256, 0, stream>>>(x1, x2, Mpad);
  k2_mu   <<<(B_*H_*(Q_/16)*(C_/16)) / 8, 256, 0, stream>>>(Mpad, U, Y);
  k3_gemm <<<B_*H_*(Q_/128)*(Q_/128), 256, 0, stream>>>(Mpad, Y, rowmx, colmx, kmin);
  k4_alpha<<<(B_*H_*Q_) / 256, 256, 0, stream>>>(rowmx, colmx, kmin, alpha);
  k5_agg  <<<B_*H_*2, 256, 0, stream>>>(Mpad, alpha, (float*)d_out);
}